// EMAVectorQuantizer_6854767804648
// MI455X (gfx1250) — compile-verified
//
#include <hip/hip_runtime.h>
#include <math.h>

// ---------------------------------------------------------------------------
// VQ-VAE vector quantizer for MI455X (gfx1250, wave32).
//   inputs: [16, 64, 64, 64] f32  (B, C, H, W), C == embedding dim = 64
//   weight: [1024, 64] f32        (K codewords)
// Outputs (concatenated f32 in d_out):
//   quantized [16,64,64,64] | loss | perplexity | indices [65536] | one-hot [65536,1024]
// Distance argmin done as argmax of (x.w - 0.5*||w||^2) via fp32 WMMA 16x16x4.
// ---------------------------------------------------------------------------

typedef __attribute__((ext_vector_type(2))) float v2f;
typedef __attribute__((ext_vector_type(8))) float v8f;

#define NUM_K   1024
#define CDIM    64
#define NVEC    65536            // 16*64*64 vectors
#define HW      4096             // 64*64
#define CHW     262144           // 64*4096
#define NTILES  4096             // NVEC / 16

// d_out layout (float element offsets)
#define Q_OFF    0u
#define LOSS_OFF 4194304u
#define PERP_OFF 4194305u
#define IDX_OFF  4194306u
#define ENC_OFF  4259842ull

// ---------------------------------------------------------------------------
// Kernel 0: precompute 0.5*||w_k||^2 and zero the histogram.
// ---------------------------------------------------------------------------
__global__ void vq_prep(const float* __restrict__ w,
                        float* __restrict__ wnorm,
                        unsigned* __restrict__ counts) {
  int k = blockIdx.x * blockDim.x + threadIdx.x;
  if (k < NUM_K) {
    float s = 0.f;
    #pragma unroll 8
    for (int c = 0; c < CDIM; ++c) { float v = w[k * CDIM + c]; s += v * v; }
    wnorm[k] = 0.5f * s;
    counts[k] = 0u;
  }
}

// ---------------------------------------------------------------------------
// Kernel 1: per-wave 16-row tile: WMMA scores vs all 1024 codewords, argmax,
// then write quantized / indices / histogram / loss partial / one-hot rows.
// ---------------------------------------------------------------------------
__global__ __launch_bounds__(256) void vq_main(const float* __restrict__ x,
                                               const float* __restrict__ w,
                                               const float* __restrict__ wnorm,
                                               float* __restrict__ out,
                                               unsigned* __restrict__ counts,
                                               float* __restrict__ lossp) {
  __shared__ int sIdx[8][16];

  const int lane = threadIdx.x & 31;
  const int wid  = threadIdx.x >> 5;
  const int tile = blockIdx.x * 8 + wid;       // 4096 tiles total
  const int n0   = tile * 16;                  // first flat vector of tile
  const int b    = n0 >> 12;                   // batch index (n0 multiple of 16)
  const int hw0  = n0 & 4095;                  // h*64+w base (rows share b,h)
  const int in_base = b * CHW + hw0;           // x[in_base + c*HW + m]
  const int m  = lane & 15;                    // A row / B column within tile
  const int hi = lane >> 4;                    // half-wave selects K sub-pair

  // --- A fragments: 16x64 tile as 16 steps of K=4 (ISA 32-bit A layout:
  //     lanes 0-15 M=0..15 hold K=0,1; lanes 16-31 hold K=2,3) ---
  v2f a[16];
  #pragma unroll
  for (int kk = 0; kk < 16; ++kk) {
    const int c0 = 4 * kk + 2 * hi;
    v2f av;
    av.x = x[in_base + c0 * HW + m];
    av.y = x[in_base + (c0 + 1) * HW + m];
    a[kk] = av;
  }

  // --- scan 64 codeword tiles; track best (score, index) per row slot ---
  float best[8];
  int   bidx[8];
  #pragma unroll
  for (int r = 0; r < 8; ++r) { best[r] = -3.4e38f; bidx[r] = 0; }

  for (int j = 0; j < 64; ++j) {
    const int col = j * 16 + m;                       // global codeword column
    const float* wr = w + col * CDIM + 2 * hi;        // 8B-aligned
    v8f acc = {0.f, 0.f, 0.f, 0.f, 0.f, 0.f, 0.f, 0.f};
    #pragma unroll
    for (int kk = 0; kk < 16; ++kk) {
      v2f bf = *(const v2f*)(wr + 4 * kk);
      acc = __builtin_amdgcn_wmma_f32_16x16x4_f32(
          /*neg_a=*/false, a[kk], /*neg_b=*/false, bf,
          /*c_mod=*/(short)0, acc, /*reuse_a=*/false, /*reuse_b=*/false);
    }
    const float wn = wnorm[col];
    #pragma unroll
    for (int r = 0; r < 8; ++r) {
      const float s = acc[r] - wn;                    // -0.5*distance + const
      if (s > best[r]) { best[r] = s; bidx[r] = col; } // strict > : first max
    }
  }

  // --- argmax reduce across the 16 lanes sharing each row group;
  //     tie-break on smaller codeword index to match jnp.argmin ---
  #pragma unroll
  for (int r = 0; r < 8; ++r) {
    float s = best[r]; int id = bidx[r];
    #pragma unroll
    for (int mask = 1; mask < 16; mask <<= 1) {
      float os = __shfl_xor(s, mask, 32);
      int   oi = __shfl_xor(id, mask, 32);
      if (os > s || (os == s && oi < id)) { s = os; id = oi; }
    }
    best[r] = s; bidx[r] = id;
  }

  // lane 0 owns rows 0..7, lane 16 owns rows 8..15
  if (lane == 0 || lane == 16) {
    const int rowbase = hi * 8;
    #pragma unroll
    for (int r = 0; r < 8; ++r) {
      const int k = bidx[r];
      sIdx[wid][rowbase + r] = k;
      out[IDX_OFF + (unsigned)(n0 + rowbase + r)] = (float)k;
      atomicAdd(&counts[k], 1u);                      // integer: deterministic
    }
  }
  __syncthreads();

  // --- quantized output (same [B,C,H,W] addressing as input) + loss partial ---
  const int myk = sIdx[wid][m];
  const float* wk = w + myk * CDIM;
  float lsum = 0.f;
  #pragma unroll
  for (int it = 0; it < 32; ++it) {
    const int c   = hi + 2 * it;                      // covers c = 0..63
    const int off = in_base + c * HW + m;
    const float wq = wk[c];
    const float xi = x[off];
    out[Q_OFF + (unsigned)off] = wq;
    const float d = wq - xi;
    lsum += d * d;
  }
  #pragma unroll
  for (int mask = 1; mask < 32; mask <<= 1) lsum += __shfl_xor(lsum, mask, 32);
  if (lane == 0) lossp[tile] = lsum;                  // one writer per slot

  // --- one-hot encodings: 16 rows x 1024 cols, coalesced float2 stores ---
  float* enc = out + ENC_OFF + (unsigned long long)n0 * NUM_K;
  for (int r = 0; r < 16; ++r) {
    const int kr = sIdx[wid][r];
    float* rowp = enc + r * NUM_K;
    #pragma unroll
    for (int it = 0; it < 16; ++it) {
      const int c2 = it * 64 + lane * 2;              // 64 floats per iter
      float2 v;
      v.x = (c2     == kr) ? 1.f : 0.f;
      v.y = (c2 + 1 == kr) ? 1.f : 0.f;
      *(float2*)(rowp + c2) = v;
    }
  }
}

// ---------------------------------------------------------------------------
// Kernel 2: deterministic fixed-order reductions -> loss & perplexity scalars.
// ---------------------------------------------------------------------------
__global__ void vq_finalize(const unsigned* __restrict__ counts,
                            const float* __restrict__ lossp,
                            float* __restrict__ out) {
  __shared__ float red[256];
  const int t = threadIdx.x;

  float ls = 0.f;
  for (int i = t; i < NTILES; i += 256) ls += lossp[i];
  red[t] = ls; __syncthreads();
  for (int s = 128; s > 0; s >>= 1) { if (t < s) red[t] += red[t + s]; __syncthreads(); }
  if (t == 0) out[LOSS_OFF] = 0.25f * red[0] / 4194304.0f;   // mean over B*C*H*W
  __syncthreads();

  float hs = 0.f;
  for (int k = t; k < NUM_K; k += 256) {
    const float p = (float)counts[k] * (1.0f / (float)NVEC);
    hs += p * logf(p + 1e-10f);
  }
  red[t] = hs; __syncthreads();
  for (int s = 128; s > 0; s >>= 1) { if (t < s) red[t] += red[t + s]; __syncthreads(); }
  if (t == 0) out[PERP_OFF] = expf(-red[0]);
}

// ---------------------------------------------------------------------------
extern "C" void kernel_launch(void* const* d_in, const int* in_sizes, int n_in,
                              void* d_out, int out_size, void* d_ws, size_t ws_size,
                              hipStream_t stream) {
  const float* x = (const float*)d_in[0];   // [16,64,64,64]
  const float* w = (const float*)d_in[1];   // [1024,64]
  float* out = (float*)d_out;

  // workspace: 4 KB wnorm | 4 KB counts | 16 KB loss partials  (24 KB total)
  float*    wnorm  = (float*)d_ws;
  unsigned* counts = (unsigned*)((char*)d_ws + 4096);
  float*    lossp  = (float*)((char*)d_ws + 8192);

  vq_prep<<<4, 256, 0, stream>>>(w, wnorm, counts);
  vq_main<<<512, 256, 0, stream>>>(x, w, wnorm, out, counts, lossp);
  vq_finalize<<<1, 256, 0, stream>>>(counts, lossp, out);
}